// DetectionLoss_58497454571628
// MI455X (gfx1250) — compile-verified
//
#include <hip/hip_runtime.h>
#include <math.h>

#define NUM_CLASSES 24
#define GRIDN 14
#define HW 196            // 14*14
#define CTOT 34           // 24 + 2 + 8 channels
#define BATCH 8192
#define CELLF 32.0f       // 448/14
#define IMGF 448.0f
#define COORD_SCALE 5.0f

typedef __attribute__((ext_vector_type(2))) float v2f;
typedef __attribute__((ext_vector_type(8))) float v8f;

// ---------------- K0: init scratch + output ----------------
__global__ void k0_zero(float* __restrict__ out, unsigned long long* __restrict__ packed) {
    int t = threadIdx.x;
    if (t == 0) out[0] = 0.0f;
    if (t < HW) packed[t] = 0ull;
}

// ---------------- K1: per-batch gather + coord/class loss + tc scatter ----------------
__global__ void k1_perbatch(const float* __restrict__ output,
                            const float* __restrict__ coords,
                            const int*  __restrict__ label,
                            float* __restrict__ partial,
                            int*   __restrict__ resp,
                            unsigned long long* __restrict__ packed) {
    int b = blockIdx.x * blockDim.x + threadIdx.x;
    if (b >= BATCH) return;

    float x  = coords[b * 4 + 0];
    float y  = coords[b * 4 + 1];
    float tw = coords[b * 4 + 2];
    float th = coords[b * 4 + 3];

    float txc = x + tw * 0.5f;
    float tyc = y + th * 0.5f;
    int cx = (int)floorf(txc / CELLF);
    int cy = (int)floorf(tyc / CELLF);
    float tx = txc - (float)cx * CELLF;
    float ty = tyc - (float)cy * CELLF;

    const float* base = output + (size_t)b * (CTOT * HW);

    // pred_bbox.reshape(b,14,14,8)[b,cx,cy] == 8 contiguous floats in memory:
    // flat offset inside the contiguous 8*196 slice is cx*112 + cy*8.
    const float* boxp = base + 26 * HW + cx * 112 + cy * 8;   // 32B aligned
    float bx[8];
#pragma unroll
    for (int k = 0; k < 8; ++k) bx[k] = boxp[k];

    float tx1 = tx - tw * 0.5f, tx2 = tx + tw * 0.5f;
    float ty1 = ty - th * 0.5f, ty2 = ty + th * 0.5f;
    float areaA = tw * th;

    float iou[2];
#pragma unroll
    for (int j = 0; j < 2; ++j) {
        float pxc = bx[j * 4 + 0], pyc = bx[j * 4 + 1];
        float pwc = bx[j * 4 + 2], phc = bx[j * 4 + 3];
        float px1 = pxc - pwc * 0.5f, px2 = pxc + pwc * 0.5f;
        float py1 = pyc - phc * 0.5f, py2 = pyc + phc * 0.5f;
        float dx = fmaxf(0.0f, fminf(tx2, px2) - fminf(IMGF, fmaxf(tx1, px1)));
        float dy = fmaxf(0.0f, fminf(ty2, py2) - fminf(IMGF, fmaxf(ty1, py1)));
        float inter = dx * dy;
        float areaB = pwc * phc;
        float v = inter / (areaA + areaB - inter);
        iou[j] = (v < 1e-6f) ? 0.0f : v;
    }
    int   r    = (iou[1] > iou[0]) ? 1 : 0;           // argmax, first wins ties
    float conf = fmaxf(iou[0], iou[1]);

    // coordinate loss with the responsible box
    float px = bx[r * 4 + 0], py = bx[r * 4 + 1];
    float pw = bx[r * 4 + 2], ph = bx[r * 4 + 3];
    pw = fminf(fmaxf(pw, 0.0f), IMGF);
    ph = fminf(fmaxf(ph, 0.0f), IMGF);
    float dtx = tx - px, dty = ty - py;
    float dsw = sqrtf(tw) - sqrtf(pw);
    float dsh = sqrtf(th) - sqrtf(ph);
    float loss = COORD_SCALE * (dtx * dtx + dty * dty + dsw * dsw + dsh * dsh);

    // class loss: softmax over 24 logits at the object cell only
    int pos = cx * GRIDN + cy;
    float cls[NUM_CLASSES];
    float m = -INFINITY;
#pragma unroll
    for (int c = 0; c < NUM_CLASSES; ++c) {
        cls[c] = base[c * HW + pos];
        m = fmaxf(m, cls[c]);
    }
    float s = 0.0f;
#pragma unroll
    for (int c = 0; c < NUM_CLASSES; ++c) { cls[c] = __expf(cls[c] - m); s += cls[c]; }
    float inv = 1.0f / s;
    int lab = label[b] - 1;
    float ce = 0.0f;
#pragma unroll
    for (int c = 0; c < NUM_CLASSES; ++c) {
        float d = cls[c] * inv - ((c == lab) ? 1.0f : 0.0f);
        ce += d * d;
    }
    loss += ce;

    partial[b] = loss;
    resp[b]    = r;

    // deterministic last-writer-wins scatter of confIoU into shared tc grid
    unsigned long long key =
        ((unsigned long long)(unsigned)(b + 1) << 32) | (unsigned long long)__float_as_uint(conf);
    atomicMax(&packed[pos], key);
}

// ---------------- K2: object loss, one wave32 per batch, WMMA wave-reduction ----------------
__global__ void k2_obj(const float* __restrict__ output,
                       const float* __restrict__ partial,
                       const int*   __restrict__ resp,
                       const unsigned long long* __restrict__ packed,
                       float* __restrict__ out) {
    __shared__ float tc[HW];
    __shared__ float wsum[8];

    int tid  = threadIdx.x;
    int lane = tid & 31;
    int wave = tid >> 5;
    int b    = blockIdx.x * 8 + wave;

    int r = resp[b];
    const float* rc = output + (size_t)b * (CTOT * HW) + (24 + r) * HW;

    // overlap: prefetch the gathered 784B confidence row while tc fills LDS
    if (lane == 0) {
        __builtin_prefetch(rc, 0, 3);
        __builtin_prefetch(rc + 64, 0, 3);
        __builtin_prefetch(rc + 128, 0, 3);
    }

    if (tid < HW) {
        unsigned long long pk = packed[tid];
        tc[tid] = (pk >> 32) ? __uint_as_float((unsigned)pk) : 0.0f;
    }
    __syncthreads();

    float acc = 0.0f;
    for (int g = lane; g < HW; g += 32) {
        float d = tc[g] - rc[g];
        acc += d * d;
    }
    if (lane == 0) acc += partial[b];

    // Wave32 reduction on the matrix pipe: D = A(16x4) * ones(4x16).
    // A layout: lane L<16 -> (M=L,K=0) in a[0]; lane L>=16 -> (M=L-16,K=2).
    // With a[1]=0: D[m,n] = acc[m] + acc[m+16] for every n.
    v2f a;    a[0] = acc;  a[1] = 0.0f;
    v2f ones; ones[0] = 1.0f; ones[1] = 1.0f;
    v8f c = {};
    c = __builtin_amdgcn_wmma_f32_16x16x4_f32(false, a, false, ones,
                                              (short)0, c, false, false);
    // lane0 holds column 0 of rows 0..7, lane16 rows 8..15
    float ssum = c[0] + c[1] + c[2] + c[3] + c[4] + c[5] + c[6] + c[7];
    ssum += __shfl_xor(ssum, 16, 32);            // lane0 now has full 32-lane sum
    if (lane == 0) wsum[wave] = ssum;
    __syncthreads();

    if (tid == 0) {
        float t = 0.0f;
#pragma unroll
        for (int w = 0; w < 8; ++w) t += wsum[w];
        atomicAdd(out, t);
    }
}

// ---------------- launcher ----------------
extern "C" void kernel_launch(void* const* d_in, const int* in_sizes, int n_in,
                              void* d_out, int out_size, void* d_ws, size_t ws_size,
                              hipStream_t stream) {
    const float* output = (const float*)d_in[0];
    const float* coords = (const float*)d_in[1];
    const int*   label  = (const int*)d_in[2];
    float* out = (float*)d_out;

    unsigned long long* packed = (unsigned long long*)d_ws;            // 196 * 8B
    float* partial = (float*)((char*)d_ws + 2048);                     // 8192 * 4B
    int*   resp    = (int*)((char*)d_ws + 2048 + BATCH * 4);           // 8192 * 4B

    k0_zero<<<dim3(1), dim3(256), 0, stream>>>(out, packed);
    k1_perbatch<<<dim3(BATCH / 256), dim3(256), 0, stream>>>(output, coords, label,
                                                             partial, resp, packed);
    k2_obj<<<dim3(BATCH / 8), dim3(256), 0, stream>>>(output, partial, resp, packed, out);
}